// GraphConv_6150393168669
// MI455X (gfx1250) — compile-verified
//
#include <hip/hip_runtime.h>
#include <hip/hip_bf16.h>

#define D 128
#define ALPHA 0.1f
#define BETA 0.22314355131420976f  /* log(1.25) */
#define C0 ((1.0f - ALPHA) * (1.0f - BETA))
#define C1 (ALPHA * (1.0f - BETA))

typedef __attribute__((ext_vector_type(2))) float v2f;
typedef __attribute__((ext_vector_type(8))) float v8f;

// ---- 1) deg = 1 (self-loop contribution) ----
__global__ void k_deg_init(float* __restrict__ deg, int N) {
    int i = blockIdx.x * blockDim.x + threadIdx.x;
    if (i < N) deg[i] = 1.0f;
}

// ---- 2) deg[col[e]] += 1 ----
__global__ void k_deg_count(const int* __restrict__ col, float* __restrict__ deg, int E) {
    int e = blockIdx.x * blockDim.x + threadIdx.x;
    if (e < E) atomicAdd(&deg[col[e]], 1.0f);
}

// ---- 3) deg_inv = 1 / clamp(deg, 1) ----
__global__ void k_deg_inv(float* __restrict__ deg, int N) {
    int i = blockIdx.x * blockDim.x + threadIdx.x;
    if (i < N) deg[i] = 1.0f / fmaxf(deg[i], 1.0f);
}

// ---- 4) x_l init with self-loop term: ew(self) = deg_inv + DIAG_LAMBDA*deg_inv = 2*deg_inv ----
__global__ void k_xl_init(const float* __restrict__ x, const float* __restrict__ deg_inv,
                          float* __restrict__ xl, int total /* = N*D */) {
    int idx = blockIdx.x * blockDim.x + threadIdx.x;  // one float each
    if (idx < total) {
        int row = idx >> 7;  // /D
        xl[idx] = 2.0f * deg_inv[row] * x[idx];
    }
}

// ---- 5) edge scatter: one wave32 per edge, 4 floats per lane ----
__global__ void k_scatter(const int* __restrict__ row, const int* __restrict__ col,
                          const float* __restrict__ x, const float* __restrict__ deg_inv,
                          float* __restrict__ xl, int E) {
    int warp = (blockIdx.x * blockDim.x + threadIdx.x) >> 5;
    int lane = threadIdx.x & 31;
    if (warp >= E) return;
    int r = row[warp];
    int c = col[warp];
    float w = deg_inv[c];
    const float* xs = x + (long long)r * D;
    float* xd = xl + (long long)c * D;
    #pragma unroll
    for (int k = 0; k < 4; ++k) {
        int d = lane + 32 * k;
        atomicAdd(&xd[d], w * xs[d]);
    }
}

// ---- 6) fused WMMA GEMM + combine + ReLU ----
// One wave computes a 16x16 output tile of: acc = x_l@W1 + x_0@W2 (K=128 each),
// then out = relu(C0*x_l + BETA*acc + C1*x_0 + bias).
__global__ void k_gemm(const float* __restrict__ xl, const float* __restrict__ x0,
                       const float* __restrict__ W1, const float* __restrict__ W2,
                       const float* __restrict__ bias, float* __restrict__ out, int N) {
    const int warp = threadIdx.x >> 5;       // 8 waves/block -> 8 column tiles
    const int lane = threadIdx.x & 31;
    const int n    = lane & 15;              // N index (B/C/D) and M index (A)
    const int half = lane >> 4;              // selects K pair {0,1} vs {2,3}
    const int i0   = blockIdx.x * 16;        // row block
    const int j0   = warp * 16;              // col block (D/16 == 8 == waves/block)
    if (i0 >= N) return;

    const int arow = min(i0 + n, N - 1);     // clamp: EXEC must stay all-1 for WMMA
    const long long abase = (long long)arow * D;

    v8f acc = {};
    // acc += x_l @ W1
    for (int k0 = 0; k0 < D; k0 += 4) {
        int ka = k0 + half * 2;
        v2f a, b;
        a.x = xl[abase + ka];
        a.y = xl[abase + ka + 1];
        b.x = W1[(long long)ka * D + j0 + n];
        b.y = W1[(long long)(ka + 1) * D + j0 + n];
        acc = __builtin_amdgcn_wmma_f32_16x16x4_f32(false, a, false, b,
                                                    (short)0, acc, false, false);
    }
    // acc += x_0 @ W2
    for (int k0 = 0; k0 < D; k0 += 4) {
        int ka = k0 + half * 2;
        v2f a, b;
        a.x = x0[abase + ka];
        a.y = x0[abase + ka + 1];
        b.x = W2[(long long)ka * D + j0 + n];
        b.y = W2[(long long)(ka + 1) * D + j0 + n];
        acc = __builtin_amdgcn_wmma_f32_16x16x4_f32(false, a, false, b,
                                                    (short)0, acc, false, false);
    }

    const float bn = bias[j0 + n];
    #pragma unroll
    for (int r = 0; r < 8; ++r) {
        int m  = r + 8 * half;               // C/D layout: VGPR r holds M=r / M=8+r
        int gi = i0 + m;
        if (gi < N) {
            long long o = (long long)gi * D + j0 + n;
            float v = C0 * xl[o] + BETA * acc[r] + C1 * x0[o] + bn;
            out[o] = fmaxf(v, 0.0f);
        }
    }
}

extern "C" void kernel_launch(void* const* d_in, const int* in_sizes, int n_in,
                              void* d_out, int out_size, void* d_ws, size_t ws_size,
                              hipStream_t stream) {
    const float* x    = (const float*)d_in[0];
    const float* x_0  = (const float*)d_in[1];
    const float* W1   = (const float*)d_in[2];
    const float* W2   = (const float*)d_in[3];
    const float* bias = (const float*)d_in[4];
    const int*   ei   = (const int*)d_in[5];

    const int ND = in_sizes[0];
    const int N  = ND / D;
    const int E  = in_sizes[5] / 2;
    const int* row = ei;
    const int* col = ei + E;

    float* deg_inv = (float*)d_ws;           // N floats (deg, then inverted in place)
    float* xl      = deg_inv + N;            // N*D floats

    const int B = 256;
    k_deg_init <<<(N + B - 1) / B, B, 0, stream>>>(deg_inv, N);
    k_deg_count<<<(E + B - 1) / B, B, 0, stream>>>(col, deg_inv, E);
    k_deg_inv  <<<(N + B - 1) / B, B, 0, stream>>>(deg_inv, N);
    k_xl_init  <<<(ND + B - 1) / B, B, 0, stream>>>(x, deg_inv, xl, ND);
    k_scatter  <<<(E + 7) / 8, B, 0, stream>>>(row, col, x, deg_inv, xl, E);
    k_gemm     <<<(N + 15) / 16, B, 0, stream>>>(xl, x_0, W1, W2, bias, (float*)d_out, N);
}